// Time_Aware_Positional_Encoding_4999341932716
// MI455X (gfx1250) — compile-verified
//
#include <hip/hip_runtime.h>
#include <stdint.h>

// ---------------------------------------------------------------------------
// Time-aware positional encoding, closed form:
//   start = L - data_size[b]
//   mean  = (time[b,L-1] - time[b,start]) / max(data_size-1, 1)
//   pos(j) = j >= start ? (j - start + 1) + (time[b,j] - time[b,start])/mean : 0
//   out[b,j,2k]   = x[b,j,2k]   + sin(pos * div_term[k])
//   out[b,j,2k+1] = x[b,j,2k+1] + cos(pos * div_term[k])
// Memory-bound (~270 MB): stream x through LDS with CDNA5 async global->LDS
// copies (ASYNCcnt), double-buffered per wave; nontemporal stores for out.
// ---------------------------------------------------------------------------

namespace {
constexpr int kB            = 128;
constexpr int kL            = 2048;
constexpr int kD            = 128;
constexpr int kRowsPerChunk = 16;                         // 16 rows * 512B = 8KB per buffer
constexpr int kChunks       = 4;                          // chunks per block
constexpr int kRowsPerBlock = kRowsPerChunk * kChunks;    // 64 rows
constexpr int kThreads      = 256;                        // 8 waves
constexpr int kF4PerChunk   = kRowsPerChunk * kD / 4;     // 512 float4 per chunk
constexpr int kF4PerThread  = kF4PerChunk / kThreads;     // 2 per thread per chunk
constexpr int kSegsPerBatch = kL / kRowsPerBlock;         // 32 blocks per batch
}

typedef __attribute__((ext_vector_type(4))) float v4f;
typedef __attribute__((ext_vector_type(4))) int   v4i;
typedef __attribute__((address_space(1))) v4i Gv4i;   // global int4
typedef __attribute__((address_space(3))) v4i Lv4i;   // LDS int4

__device__ __forceinline__ void async_copy_b128(const void* gsrc, float* lds_dst) {
#if __has_builtin(__builtin_amdgcn_global_load_async_to_lds_b128)
  __builtin_amdgcn_global_load_async_to_lds_b128(
      (Gv4i*)gsrc, (Lv4i*)lds_dst, /*offset=*/0, /*cpol=*/0);
#else
  unsigned lds_off = (unsigned)(uintptr_t)(Lv4i*)lds_dst;
  unsigned long long ga = (unsigned long long)(uintptr_t)gsrc;
  asm volatile("global_load_async_to_lds_b128 %0, %1, off"
               :: "v"(lds_off), "v"(ga) : "memory");
#endif
}

__global__ __launch_bounds__(kThreads)
void tape_pe_kernel(const float* __restrict__ x,
                    const float* __restrict__ tm,
                    const float* __restrict__ div_term,
                    const int*   __restrict__ data_size,
                    float*       __restrict__ out)
{
  __shared__ float xbuf[2][kF4PerChunk * 4];   // 2 x 8KB double buffer

  const int blk  = blockIdx.x;
  const int b    = blk / kSegsPerBatch;
  const int seg  = blk % kSegsPerBatch;
  const int row0 = seg * kRowsPerBlock;

  // Per-batch scalars (computed redundantly per thread; loads broadcast in L0/L2).
  const int   ds      = data_size[b];
  const int   start   = kL - ds;
  const float tstart  = tm[(size_t)b * kL + start];
  const float tlast   = tm[(size_t)b * kL + (kL - 1)];
  const float denom   = (float)((ds > 1) ? (ds - 1) : 1);
  const float invmean = denom / (tlast - tstart);

  const size_t  rowBase = ((size_t)b * kL + row0) * kD;
  const v4f*    xg      = (const v4f*)(x + rowBase);
  v4f*          og      = (v4f*)(out + rowBase);
  const float*  tmRow   = tm + (size_t)b * kL + row0;

  const int tid = threadIdx.x;
  const int c4  = tid & 31;      // float4 column within a row (same for both slots)
  const int rA  = tid >> 5;      // first row-in-chunk handled by this thread

  const float dt0 = div_term[c4 * 2 + 0];
  const float dt1 = div_term[c4 * 2 + 1];

  auto issue_chunk = [&](int c, int bi) {
#pragma unroll
    for (int t = 0; t < kF4PerThread; ++t) {
      const int f = tid + t * kThreads;                 // float4 slot in chunk
      async_copy_b128(xg + (size_t)c * kF4PerChunk + f, &xbuf[bi][f * 4]);
    }
  };

  auto process_chunk = [&](int c, int bi) {
    const int jbase = row0 + c * kRowsPerChunk;
#pragma unroll
    for (int t = 0; t < kF4PerThread; ++t) {
      const int f = tid + t * kThreads;
      const int r = rA + t * (kThreads >> 5);           // row within chunk
      const int j = jbase + r;

      const float tj  = tmRow[c * kRowsPerChunk + r];   // broadcast within wave
      const float pos = (j >= start)
          ? ((float)(j - start + 1) + (tj - tstart) * invmean)
          : 0.0f;
      const float a0 = pos * dt0;
      const float a1 = pos * dt1;

      const v4f xin = *(const v4f*)&xbuf[bi][f * 4];    // ds_load_b128
      v4f o;
      o.x = xin.x + __sinf(a0);
      o.y = xin.y + __cosf(a0);
      o.z = xin.z + __sinf(a1);
      o.w = xin.w + __cosf(a1);
      __builtin_nontemporal_store(o, og + (size_t)c * kF4PerChunk + f);
    }
  };

  // Software pipeline: async copies of chunk c+1 overlap sin/cos of chunk c.
  issue_chunk(0, 0);
#pragma unroll
  for (int c = 0; c < kChunks; ++c) {
    if (c + 1 < kChunks) {
      asm volatile("s_wait_dscnt 0x0" ::: "memory");    // LDS reads of prev chunk drained
      issue_chunk(c + 1, (c + 1) & 1);
      asm volatile("s_wait_asynccnt 0x2" ::: "memory"); // current chunk resident, next in flight
    } else {
      asm volatile("s_wait_asynccnt 0x0" ::: "memory");
    }
    process_chunk(c, c & 1);
  }
}

extern "C" void kernel_launch(void* const* d_in, const int* in_sizes, int n_in,
                              void* d_out, int out_size, void* d_ws, size_t ws_size,
                              hipStream_t stream) {
  (void)in_sizes; (void)n_in; (void)out_size; (void)d_ws; (void)ws_size;
  const float* x         = (const float*)d_in[0];
  const float* tm        = (const float*)d_in[1];
  const float* div_term  = (const float*)d_in[2];
  const int*   data_size = (const int*)d_in[3];
  float*       out       = (float*)d_out;

  const int grid = kB * kSegsPerBatch;   // 4096 blocks
  tape_pe_kernel<<<grid, kThreads, 0, stream>>>(x, tm, div_term, data_size, out);
}